// HexagonalSensor_39152921870458
// MI455X (gfx1250) — compile-verified
//
#include <hip/hip_runtime.h>
#include <math.h>

// ---------------------------------------------------------------------------
// HexagonalSensor scatter-reduce for MI455X (gfx1250, wave32)
//   d_in[0]=x f32[N], d_in[1]=y f32[N], d_in[2]=values f32[N],
//   d_in[3]=lookup i32[Q*R], d_in[4]=hex_size f32[1], d_in[5]=q_offset f32[1],
//   d_in[6]=r_offset f32[1], d_in[7]=q_min i32[1], d_in[8]=r_min i32[1],
//   d_in[9]=n_pixels i32[1].  d_out = f32[n_pixels].
// ---------------------------------------------------------------------------

#define BLOCK 256
#define GRID  768

#if __has_builtin(__builtin_amdgcn_global_load_async_to_lds_b128) && \
    __has_builtin(__builtin_amdgcn_s_wait_asynccnt)
#define HAVE_ASYNC_LDS 1
#else
#define HAVE_ASYNC_LDS 0
#endif

// Typed pointers for the async global->LDS DMA builtin. Compiler diagnostic
// showed the expected param type: 'int __vector_size__(16) __device__ *'
// (i.e. int4 in the global address space); LDS side mirrors it in AS(3).
typedef int v4i_vec __attribute__((vector_size(16)));
typedef __attribute__((address_space(1))) v4i_vec* g_v4i_ptr;
typedef __attribute__((address_space(3))) v4i_vec* l_v4i_ptr;

__global__ void hex_zero_out(float* __restrict__ out, int n) {
  int i = blockIdx.x * blockDim.x + threadIdx.x;
  if (i < n) out[i] = 0.0f;
}

__device__ __forceinline__ void hex_accum(
    float x, float y, float v,
    float inv_h, float q_off, float r_off,
    int q_min, int r_min, int Q, int R,
    const int* __restrict__ lut, float* __restrict__ hist)
{
  const float C_Q  = 0.57735026918962576451f;  // sqrt(3)/3
  const float C_13 = 0.33333333333333333333f;
  const float C_23 = 0.66666666666666666667f;

  float q = (C_Q * x - y * C_13) * inv_h - q_off;
  float r = (C_23 * y) * inv_h - r_off;
  float s = -q - r;
  float qr = rintf(q), rr = rintf(r), sr = rintf(s);   // round-half-even
  float qd = fabsf(qr - q), rd = fabsf(rr - r), sd = fabsf(sr - s);
  // Reference applies both fix-ups from the *original* rounded values.
  float q2 = ((qd > rd) && (qd > sd)) ? (-rr - sr) : qr;
  float r2 = ((rd > qd) && (rd > sd)) ? (-qr - sr) : rr;

  int qi = (int)q2 - q_min;
  int ri = (int)r2 - r_min;
  if ((unsigned)qi < (unsigned)Q && (unsigned)ri < (unsigned)R) {
    int p = lut[qi * R + ri];          // LDS read (ds_load_b32)
    if (p >= 0) atomicAdd(&hist[p], v); // LDS float atomic (ds_add_f32)
  }
}

__global__ __launch_bounds__(BLOCK) void hex_hist_kernel(
    const float* __restrict__ xs, const float* __restrict__ ys,
    const float* __restrict__ vs, const int* __restrict__ lut_g,
    const float* __restrict__ hs_p, const float* __restrict__ qo_p,
    const float* __restrict__ ro_p, const int* __restrict__ qmin_p,
    const int* __restrict__ rmin_p,
    int Q, int R, int npix, int n, float* __restrict__ out)
{
  extern __shared__ float smem[];
  float* hist = smem;
  const int lut_off = (npix + 3) & ~3;    // keep LUT 16B-aligned in LDS
  int* lut = (int*)(smem + lut_off);
  const int QR = Q * R;

  // --- Stage the lookup table into LDS. Prefer the gfx1250 async DMA path
  //     (global_load_async_to_lds_b128, tracked by ASYNCcnt) so the copy
  //     overlaps with zeroing the LDS histogram below.
#if HAVE_ASYNC_LDS
  {
    int nchunk = QR >> 2;                 // 16-byte chunks
    for (int c = threadIdx.x; c < nchunk; c += BLOCK) {
      __builtin_amdgcn_global_load_async_to_lds_b128(
          (g_v4i_ptr)(lut_g + 4 * c),
          (l_v4i_ptr)(lut + 4 * c),
          0, 0);
    }
    for (int i = (QR & ~3) + (int)threadIdx.x; i < QR; i += BLOCK)
      lut[i] = lut_g[i];                  // tail (QR not multiple of 4)
  }
#else
  for (int i = threadIdx.x; i < QR; i += BLOCK) lut[i] = lut_g[i];
#endif

  // Zero the per-block LDS histogram while the async copy is in flight.
  for (int i = threadIdx.x; i < npix; i += BLOCK) hist[i] = 0.0f;

#if HAVE_ASYNC_LDS
  __builtin_amdgcn_s_wait_asynccnt(0);
#endif
  __syncthreads();

  const float inv_h = 1.0f / *hs_p;       // uniform -> scalar loads
  const float q_off = *qo_p;
  const float r_off = *ro_p;
  const int   q_min = *qmin_p;
  const int   r_min = *rmin_p;

  const int tid    = blockIdx.x * BLOCK + threadIdx.x;
  const int stride = gridDim.x * BLOCK;
  const int n4     = n >> 2;

  const float4* x4 = (const float4*)xs;
  const float4* y4 = (const float4*)ys;
  const float4* v4 = (const float4*)vs;

  // Main stream: 128-bit coalesced loads, RT temporal hint on purpose —
  // the 192MB L2 retains most of the 240MB stream across graph replays.
  for (int i = tid; i < n4; i += stride) {
    __builtin_prefetch((const void*)(x4 + i + stride), 0, 3);
    __builtin_prefetch((const void*)(y4 + i + stride), 0, 3);
    __builtin_prefetch((const void*)(v4 + i + stride), 0, 3);
    float4 X = x4[i];
    float4 Y = y4[i];
    float4 V = v4[i];
    hex_accum(X.x, Y.x, V.x, inv_h, q_off, r_off, q_min, r_min, Q, R, lut, hist);
    hex_accum(X.y, Y.y, V.y, inv_h, q_off, r_off, q_min, r_min, Q, R, lut, hist);
    hex_accum(X.z, Y.z, V.z, inv_h, q_off, r_off, q_min, r_min, Q, R, lut, hist);
    hex_accum(X.w, Y.w, V.w, inv_h, q_off, r_off, q_min, r_min, Q, R, lut, hist);
  }
  // Scalar tail (N not multiple of 4).
  for (int i = (n4 << 2) + tid; i < n; i += stride) {
    hex_accum(xs[i], ys[i], vs[i], inv_h, q_off, r_off, q_min, r_min, Q, R, lut, hist);
  }

  __syncthreads();

  // Flush block-private histogram: hardware f32 global atomics
  // (global_atomic_add_f32), GRID*npix ~1.4M atomics total vs 20M raw.
  for (int i = threadIdx.x; i < npix; i += BLOCK) {
    float bv = hist[i];
    unsafeAtomicAdd(&out[i], bv);
  }
}

extern "C" void kernel_launch(void* const* d_in, const int* in_sizes, int n_in,
                              void* d_out, int out_size, void* d_ws, size_t ws_size,
                              hipStream_t stream) {
  (void)n_in; (void)d_ws; (void)ws_size;
  const float* xs    = (const float*)d_in[0];
  const float* ys    = (const float*)d_in[1];
  const float* vs    = (const float*)d_in[2];
  const int*   lut   = (const int*)d_in[3];
  const float* hs_p  = (const float*)d_in[4];
  const float* qo_p  = (const float*)d_in[5];
  const float* ro_p  = (const float*)d_in[6];
  const int*   qmin  = (const int*)d_in[7];
  const int*   rmin  = (const int*)d_in[8];

  const int n  = in_sizes[0];
  const int QR = in_sizes[3];
  // Lookup grid is square for this sensor (q,r both span [-R_hex, R_hex]).
  int Q = 1;
  while ((long long)(Q + 1) * (Q + 1) <= (long long)QR) ++Q;
  if (Q * Q != QR) { while (QR % Q != 0) --Q; }   // defensive fallback
  const int R = QR / Q;
  const int npix = out_size;

  float* out = (float*)d_out;

  // d_out is poisoned by the harness; zero it first (stream-ordered).
  hex_zero_out<<<(npix + BLOCK - 1) / BLOCK, BLOCK, 0, stream>>>(out, npix);

  const int lut_off = (npix + 3) & ~3;
  const size_t shmem = (size_t)(lut_off + QR) * sizeof(float);

  hex_hist_kernel<<<GRID, BLOCK, shmem, stream>>>(
      xs, ys, vs, lut, hs_p, qo_p, ro_p, qmin, rmin,
      Q, R, npix, n, out);
}